// WindowedAttentionHead_29661044146195
// MI455X (gfx1250) — compile-verified
//
#include <hip/hip_runtime.h>
#include <hip/hip_bf16.h>

typedef __attribute__((ext_vector_type(2))) float v2f;
typedef __attribute__((ext_vector_type(8))) float v8f;

#define WS        64
#define NWIN      128          // windows per batch row (8192/64)
#define WTOT      8192
#define LDSP      68           // padded LDS row stride (floats)
#define SCALE     0.125f       // 64^-0.5

__global__ __launch_bounds__(256)
void swin_attn_fp32_wmma(const float* __restrict__ q,
                         const float* __restrict__ k,
                         const float* __restrict__ v,
                         const float* __restrict__ table,
                         float* __restrict__ out_x,
                         float* __restrict__ out_attn)
{
    __shared__ float smem[3 * WS * LDSP + 128];
    float* qp_s = smem;                 // q tile, later reused for softmax probs P
    float* k_s  = smem + WS * LDSP;
    float* v_s  = smem + 2 * WS * LDSP;
    float* tab  = smem + 3 * WS * LDSP; // 127-entry rel-pos bias table

    const int tid  = threadIdx.x;
    const int wid  = blockIdx.x;        // 0..4095  (b*128 + window)
    const int bidx = wid >> 7;
    const int wlo  = wid & (NWIN - 1);  // window index within batch row
    const int lane = tid & 31;
    const int wave = tid >> 5;
    const int half = lane >> 4;         // 0: K pair 0-1, 1: K pair 2-3
    const int l15  = lane & 15;

    const size_t base = (size_t)bidx * WTOT * WS;

    // ---- stage q,k,v window into LDS via CDNA5 async global->LDS DMA;
    //      fold in cyclic shift (+32 rows mod 8192)
    {
        // LDS byte offset of smem within the workgroup allocation
        // (addrspacecast keeps the LDS offset in the low 32 bits)
        const unsigned smbase = (unsigned)(uintptr_t)smem;
        const float* qb = q + base;
        const float* kb = k + base;
        const float* vb = v + base;
        for (int kk = 0; kk < 4; ++kk) {
            int flat = kk * 1024 + tid * 4;     // 4096 floats per matrix
            int i = flat >> 6;
            int c = flat & 63;
            int g = ((wlo << 6) + i + 32) & (WTOT - 1);
            int goff = (g * WS + c) * 4;                      // global byte offset
            int ldsq = smbase + (i * LDSP + c) * 4;           // qp_s region
            int ldsk = ldsq + WS * LDSP * 4;                  // k_s region
            int ldsv = ldsq + 2 * WS * LDSP * 4;              // v_s region
            asm volatile("global_load_async_to_lds_b128 %0, %1, %2"
                         :: "v"(ldsq), "v"(goff), "s"(qb) : "memory");
            asm volatile("global_load_async_to_lds_b128 %0, %1, %2"
                         :: "v"(ldsk), "v"(goff), "s"(kb) : "memory");
            asm volatile("global_load_async_to_lds_b128 %0, %1, %2"
                         :: "v"(ldsv), "v"(goff), "s"(vb) : "memory");
        }
    }
    if (tid < 2 * WS - 1) tab[tid] = table[tid];
    asm volatile("s_wait_asynccnt 0x0" ::: "memory");  // our async writes landed
    __syncthreads();                                   // everyone's landed

    // ---- GEMM1: S = q @ k^T   (each wave: 2 of the 16 16x16 tiles)
    const int t0 = wave * 2;
    v8f acc[2];
    for (int tt = 0; tt < 2; ++tt) {
        int t  = t0 + tt;
        int ti = t >> 2, tj = t & 3;
        int m  = ti * 16 + l15;
        int n  = tj * 16 + l15;
        v8f c8 = {};
        for (int kk = 0; kk < 16; ++kk) {
            int cb = kk * 4 + half * 2;
            v2f a  = *(const v2f*)&qp_s[m * LDSP + cb];  // A[m][cb..cb+1]
            v2f bb = *(const v2f*)&k_s [n * LDSP + cb];  // B[cb][n] = k[n][cb]
            c8 = __builtin_amdgcn_wmma_f32_16x16x4_f32(
                     false, a, false, bb, (short)0, c8, false, false);
        }
        acc[tt] = c8;
    }
    __syncthreads();   // all waves done reading q LDS tile

    // ---- apply scale + rel-pos bias + shift mask, write scores into qp_s
    for (int tt = 0; tt < 2; ++tt) {
        int t  = t0 + tt;
        int ti = t >> 2, tj = t & 3;
        int j  = tj * 16 + l15;
#pragma unroll
        for (int r = 0; r < 8; ++r) {
            int i = ti * 16 + r + half * 8;
            float s = acc[tt][r] * SCALE + tab[i - j + WS - 1];
            if (wlo == NWIN - 1 && ((i < 32) != (j < 32))) s -= 100.0f;
            qp_s[i * LDSP + j] = s;
        }
    }
    __syncthreads();

    // ---- softmax: one thread per row
    if (tid < WS) {
        float* row = &qp_s[tid * LDSP];
        float mx = row[0];
        for (int j = 1; j < WS; ++j) mx = fmaxf(mx, row[j]);
        float sum = 0.0f;
        for (int j = 0; j < WS; ++j) { float e = __expf(row[j] - mx); row[j] = e; sum += e; }
        float inv = 1.0f / sum;
        for (int j = 0; j < WS; ++j) row[j] *= inv;
    }
    __syncthreads();

    // ---- store attn (second output), coalesced float4 copy
    {
        size_t abase = (size_t)wid * (WS * WS);
        for (int kk = 0; kk < 4; ++kk) {
            int flat = kk * 1024 + tid * 4;
            int i = flat >> 6, c = flat & 63;
            *(float4*)&out_attn[abase + flat] = *(const float4*)&qp_s[i * LDSP + c];
        }
    }

    // ---- GEMM2: X = P @ V, scatter through inverse roll
    for (int tt = 0; tt < 2; ++tt) {
        int t  = t0 + tt;
        int ti = t >> 2, tj = t & 3;
        int m  = ti * 16 + l15;
        int n  = tj * 16 + l15;
        v8f c8 = {};
        for (int kk = 0; kk < 16; ++kk) {
            int cb = kk * 4 + half * 2;
            v2f a = *(const v2f*)&qp_s[m * LDSP + cb];    // P[m][cb..cb+1]
            v2f bb;
            bb.x = v_s[cb * LDSP + n];                    // B[cb][n] = v[cb][n]
            bb.y = v_s[(cb + 1) * LDSP + n];
            c8 = __builtin_amdgcn_wmma_f32_16x16x4_f32(
                     false, a, false, bb, (short)0, c8, false, false);
        }
#pragma unroll
        for (int r = 0; r < 8; ++r) {
            int i = ti * 16 + r + half * 8;
            int g = ((wlo << 6) + i + 32) & (WTOT - 1);
            out_x[base + (size_t)g * WS + n] = c8[r];
        }
    }
}

extern "C" void kernel_launch(void* const* d_in, const int* in_sizes, int n_in,
                              void* d_out, int out_size, void* d_ws, size_t ws_size,
                              hipStream_t stream) {
    const float* q   = (const float*)d_in[0];
    const float* k   = (const float*)d_in[1];
    const float* v   = (const float*)d_in[2];
    const float* tab = (const float*)d_in[3];

    float* out_x    = (float*)d_out;
    float* out_attn = (float*)d_out + (size_t)32 * WTOT * WS;  // after x

    dim3 grid(32 * NWIN);   // 4096 windows
    dim3 block(256);        // 8 wave32 per window
    swin_attn_fp32_wmma<<<grid, block, 0, stream>>>(q, k, v, tab, out_x, out_attn);
}